// DigitCapsule_10462540333705
// MI455X (gfx1250) — compile-verified
//
#include <hip/hip_runtime.h>

typedef __attribute__((ext_vector_type(2))) float v2f;
typedef __attribute__((ext_vector_type(8))) float v8f;

#define CHW        2048
#define O_CAPS     32
#define D_DIM      32
#define I_DIM      16
#define BT         16
#define THREADS    256
#define CHW_PER_WG 16
#define NCHUNK     (CHW / CHW_PER_WG)   // 128

// ---- LDS layout (float offsets) ----
#define U_STRIDE   36                       // 16B-aligned rows (float4) + bank spread
#define U_OFF      0
#define U_SIZE     (BT * O_CAPS * U_STRIDE) // 18432
#define S_OFF      (U_OFF + U_SIZE)
#define S_SIZE     (BT * O_CAPS * U_STRIDE) // 18432
#define X_OFF      (S_OFF + S_SIZE)
#define X_SIZE     (BT * I_DIM)             // 256
#define T_OFF      (X_OFF + X_SIZE)
#define T_SIZE     (BT * O_CAPS)            // 512
#define MX_OFF     (T_OFF + T_SIZE)
#define DEN_OFF    (MX_OFF + BT)
#define LDS_FLOATS (DEN_OFF + BT)           // 37664 floats = 150656 B

// One fused routing pass:
//   u_hat tile = x_tile @ W  (via V_WMMA_F32_16X16X4_F32)
//   if v: t[m][o] = u_hat . v ;  c = softmax_o(t) ; else c = 1/32
//   s_partial += c * u_hat ; flushed with global f32 atomics.
__global__ __launch_bounds__(THREADS)
void caps_route_pass(const float* __restrict__ x,
                     const float* __restrict__ Wm,
                     const float* __restrict__ v,   // nullptr => uniform coupling
                     float* __restrict__ s_out)
{
    extern __shared__ float lds[];
    const int tid   = threadIdx.x;
    const int lane  = tid & 31;
    const int wave  = tid >> 5;
    const int half  = lane >> 4;      // 0 | 1
    const int ln    = lane & 15;
    const int btile = blockIdx.x;     // 0..3  (fastest -> co-scheduled, share W in L2)
    const int chunk = blockIdx.y;     // 0..127
    const int b0    = btile * BT;

    for (int i = tid; i < S_SIZE; i += THREADS) lds[S_OFF + i] = 0.0f;

    for (int it = 0; it < CHW_PER_WG; ++it) {
        const int chw = chunk * CHW_PER_WG + it;
        __syncthreads();   // protect x/u_buf from previous iteration's readers

        // stage x tile: 16 batches x 16 i
        {
            const int m = tid >> 4, i = tid & 15;
            lds[X_OFF + tid] = x[((size_t)(b0 + m) * CHW + chw) * I_DIM + i];
        }
        __syncthreads();

        // A fragments (shared across all o / d-tiles): lane l, reg r <-> A[m=l%16, k=2*(l/16)+r]
        v2f af[4];
#pragma unroll
        for (int kk = 0; kk < 4; ++kk) {
            const int i0 = 4 * kk + 2 * half;
            af[kk][0] = lds[X_OFF + ln * I_DIM + i0];
            af[kk][1] = lds[X_OFF + ln * I_DIM + i0 + 1];
        }

        // -------- Phase A: WMMA u_hat tiles --------
#pragma unroll
        for (int j = 0; j < 4; ++j) {
            const int o = wave * 4 + j;
            const size_t wbase = ((size_t)chw * O_CAPS + o) * (I_DIM * D_DIM);
            if (it + 1 < CHW_PER_WG)   // lanes fan over next chw's 2KB W slab (128B apart)
                __builtin_prefetch(&Wm[wbase + (size_t)O_CAPS * I_DIM * D_DIM
                                         + (size_t)lane * 32], 0, 3);
#pragma unroll
            for (int dt = 0; dt < 2; ++dt) {
                v8f acc = {0.f, 0.f, 0.f, 0.f, 0.f, 0.f, 0.f, 0.f};
#pragma unroll
                for (int kk = 0; kk < 4; ++kk) {
                    const int i0 = 4 * kk + 2 * half;
                    const int d  = dt * 16 + ln;
                    v2f bf;      // B[k=2*half+r, n=ln] -> W[o, i0+r, d]
                    bf[0] = Wm[wbase + (size_t)i0 * D_DIM + d];
                    bf[1] = Wm[wbase + (size_t)(i0 + 1) * D_DIM + d];
                    acc = __builtin_amdgcn_wmma_f32_16x16x4_f32(
                              false, af[kk], false, bf, (short)0, acc, false, false);
                }
#pragma unroll
                for (int r = 0; r < 8; ++r) {   // D: m = r + 8*half, n = ln
                    const int m = r + 8 * half;
                    lds[U_OFF + (m * O_CAPS + o) * U_STRIDE + dt * 16 + ln] = acc[r];
                }
            }
        }
        __syncthreads();

        // -------- Phase B: routing weights + accumulate --------
        if (v != nullptr) {
            // dots: 512 (m,o) pairs / 256 threads, float4 LDS + global loads
#pragma unroll
            for (int pp = 0; pp < 2; ++pp) {
                const int p = tid + pp * THREADS;
                const int m = p >> 5, o = p & 31;
                const float* vp = &v[(((size_t)(b0 + m)) * O_CAPS + o) * D_DIM];
                const float* up = &lds[U_OFF + (m * O_CAPS + o) * U_STRIDE];
                float t = 0.f;
#pragma unroll
                for (int q = 0; q < 8; ++q) {
                    const float4 uu = *(const float4*)(up + 4 * q);
                    const float4 vv = *(const float4*)(vp + 4 * q);
                    t += uu.x * vv.x + uu.y * vv.y + uu.z * vv.z + uu.w * vv.w;
                }
                lds[T_OFF + p] = t;
            }
            __syncthreads();
            // softmax stats: wave w owns rows 2w (lanes 0-15) and 2w+1 (lanes 16-31)
            {
                const int m = wave * 2 + half;
                const float t0 = lds[T_OFF + m * O_CAPS + ln];
                const float t1 = lds[T_OFF + m * O_CAPS + ln + 16];
                float mx = fmaxf(t0, t1);
#pragma unroll
                for (int msk = 8; msk >= 1; msk >>= 1)
                    mx = fmaxf(mx, __shfl_xor(mx, msk, 32));
                float den = __expf(t0 - mx) + __expf(t1 - mx);
#pragma unroll
                for (int msk = 8; msk >= 1; msk >>= 1)
                    den += __shfl_xor(den, msk, 32);
                if (ln == 0) { lds[MX_OFF + m] = mx; lds[DEN_OFF + m] = den; }
            }
            __syncthreads();
#pragma unroll
            for (int pp = 0; pp < 2; ++pp) {
                const int p = tid + pp * THREADS;
                const int m = p >> 5, o = p & 31;
                const float c = __expf(lds[T_OFF + p] - lds[MX_OFF + m]) / lds[DEN_OFF + m];
                float*       sp = &lds[S_OFF + (m * O_CAPS + o) * U_STRIDE];
                const float* up = &lds[U_OFF + (m * O_CAPS + o) * U_STRIDE];
#pragma unroll
                for (int q = 0; q < 8; ++q) {
                    float4 ss = *(float4*)(sp + 4 * q);
                    const float4 uu = *(const float4*)(up + 4 * q);
                    ss.x += c * uu.x;  ss.y += c * uu.y;
                    ss.z += c * uu.z;  ss.w += c * uu.w;
                    *(float4*)(sp + 4 * q) = ss;
                }
            }
        } else {
            const float c = 1.0f / 32.0f;               // softmax(0) over O=32
#pragma unroll
            for (int pp = 0; pp < 2; ++pp) {
                const int p = tid + pp * THREADS;
                const int m = p >> 5, o = p & 31;
                float*       sp = &lds[S_OFF + (m * O_CAPS + o) * U_STRIDE];
                const float* up = &lds[U_OFF + (m * O_CAPS + o) * U_STRIDE];
#pragma unroll
                for (int q = 0; q < 8; ++q) {
                    float4 ss = *(float4*)(sp + 4 * q);
                    const float4 uu = *(const float4*)(up + 4 * q);
                    ss.x += c * uu.x;  ss.y += c * uu.y;
                    ss.z += c * uu.z;  ss.w += c * uu.w;
                    *(float4*)(sp + 4 * q) = ss;
                }
            }
        }
    }
    __syncthreads();

    // flush s partials (s is only 256 KB -> L2-resident atomics)
#pragma unroll
    for (int pp = 0; pp < 2; ++pp) {
        const int p = tid + pp * THREADS;
        const int m = p >> 5, o = p & 31;
        const float* sp = &lds[S_OFF + (m * O_CAPS + o) * U_STRIDE];
        float* gp = &s_out[(((size_t)(b0 + m)) * O_CAPS + o) * D_DIM];
        for (int d = 0; d < D_DIM; ++d) atomicAdd(&gp[d], sp[d]);
    }
}

// squash over last dim (one wave32 per (b,o)); optionally emit vsum = vout + vprev
__global__ __launch_bounds__(256)
void squash_kernel(const float* __restrict__ s, float* __restrict__ vout,
                   const float* __restrict__ vprev, float* __restrict__ vsum)
{
    const int lane = threadIdx.x & 31;
    const int grp  = blockIdx.x * (blockDim.x >> 5) + (threadIdx.x >> 5);
    const int idx  = grp * 32 + lane;           // ((b*32+o)*32 + d)
    const float val = s[idx];
    float sq = val * val;
#pragma unroll
    for (int m = 16; m >= 1; m >>= 1) sq += __shfl_xor(sq, m, 32);
    const float scale = (sq / (1.0f + sq)) / (sqrtf(sq) + 1e-6f);
    const float r = val * scale;
    vout[idx] = r;
    if (vsum) vsum[idx] = r + vprev[idx];
}

extern "C" void kernel_launch(void* const* d_in, const int* in_sizes, int n_in,
                              void* d_out, int out_size, void* d_ws, size_t ws_size,
                              hipStream_t stream)
{
    const float* x  = (const float*)d_in[0];   // (64,32,8,8,16)
    const float* Wm = (const float*)d_in[1];   // (1,32,8,8,32,16,32)
    float* out = (float*)d_out;                // (64,32,32)

    const size_t N = 64 * 32 * 32;             // 65536 elements per buffer
    float* wsf = (float*)d_ws;
    float* s1  = wsf + 0 * N;
    float* s2  = wsf + 1 * N;
    float* s3  = wsf + 2 * N;
    float* v1  = wsf + 3 * N;
    float* v2  = wsf + 4 * N;
    float* v12 = wsf + 5 * N;

    hipMemsetAsync(wsf, 0, 3 * N * sizeof(float), stream);   // zero s1..s3 (capture-safe)

    dim3 grid(4, NCHUNK);
    const size_t shmem = LDS_FLOATS * sizeof(float);

    // iter 1: c uniform -> s1 ; v1 = squash(s1)
    caps_route_pass<<<grid, THREADS, shmem, stream>>>(x, Wm, nullptr, s1);
    squash_kernel<<<(64 * 32) / 8, 256, 0, stream>>>(s1, v1, nullptr, nullptr);
    // iter 2: logits b2 = u.v1 -> s2 ; v2 = squash(s2), v12 = v1 + v2
    caps_route_pass<<<grid, THREADS, shmem, stream>>>(x, Wm, v1, s2);
    squash_kernel<<<(64 * 32) / 8, 256, 0, stream>>>(s2, v2, v1, v12);
    // iter 3: logits b3 = u.(v1+v2) -> s3 ; out = squash(s3)
    caps_route_pass<<<grid, THREADS, shmem, stream>>>(x, Wm, v12, s3);
    squash_kernel<<<(64 * 32) / 8, 256, 0, stream>>>(s3, out, nullptr, nullptr);
}